// GATModel2L_76785425318471
// MI455X (gfx1250) — compile-verified
//
#include <hip/hip_runtime.h>
#include <cstddef>
#include <cstdint>

typedef float v2f __attribute__((ext_vector_type(2)));
typedef float v8f __attribute__((ext_vector_type(8)));

// ---------------------------------------------------------------------------
// helpers
// ---------------------------------------------------------------------------
__device__ __forceinline__ float lrelu(float x, float s) {
    return x > 0.0f ? x : s * x;
}

// float atomic max via sign-split integer atomics
// (global_atomic_max_i32 for v>=0, global_atomic_min_u32 for v<0).
// Target must be initialized to -inf (0xFF800000).
__device__ __forceinline__ void atomicMaxF(float* addr, float v) {
    if (v >= 0.0f) {
        atomicMax(reinterpret_cast<int*>(addr), __float_as_int(v));
    } else {
        atomicMin(reinterpret_cast<unsigned int*>(addr), __float_as_uint(v));
    }
}

// ---------------------------------------------------------------------------
// GEMM: H[n x NOUT] = X[n x K] @ W[K x NOUT], fp32 exact via WMMA 16x16x4.
// One wave computes a 16-row x NOUT tile. W is staged TRANSPOSED in LDS with
// a padded stride (K+2 floats): rows stay 8B-aligned for float2 ds_load_b64
// and the 66-float stride avoids the 64-bank wrap (lanes hit distinct banks).
// A-row index is clamped (not predicated): an out-of-range A row only feeds
// an out-of-range D row, which is never stored -> no EXEC-mask branches.
// ---------------------------------------------------------------------------
template <int K, int NOUT>
__launch_bounds__(256)
__global__ void gat_gemm_wmma(const float* __restrict__ X,
                              const float* __restrict__ W,
                              float* __restrict__ H, int n) {
    constexpr int NT  = NOUT / 16;   // 16-wide N tiles per wave
    constexpr int LDK = K + 2;       // padded (even) K-stride for W^T in LDS
    __shared__ alignas(16) float sWt[NOUT * LDK];
    for (int i = threadIdx.x; i < K * NOUT; i += 256) {
        const int k = i / NOUT;      // coalesced read of row-major W
        const int c = i - k * NOUT;
        sWt[c * LDK + k] = W[i];
    }
    __syncthreads();

    const int wave = threadIdx.x >> 5;   // wave32
    const int lane = threadIdx.x & 31;
    const int tile = blockIdx.x * 8 + wave;
    const int m0 = tile * 16;
    if (m0 >= n) return;

    const int ml = lane & 15;            // A: row M ; B: col N
    const int kh = lane >> 4;            // K half-select (0 or 1)
    int row = m0 + ml;
    row = row < n ? row : n - 1;         // clamp instead of predicate
    const float* __restrict__ xrow = X + (size_t)row * K;

    v8f acc[NT] = {};

#pragma unroll
    for (int kk = 0; kk < K; kk += 4) {
        // A 16x4 fp32: lanes 0-15 hold K=kk+0,kk+1; lanes 16-31 hold kk+2,kk+3
        const v2f a = *(const v2f*)(xrow + kk + 2 * kh);
#pragma unroll
        for (int t = 0; t < NT; ++t) {
            // B 4x16 fp32 from W^T: lane = col N, contiguous K pair -> one b64
            const v2f b = *(const v2f*)(&sWt[(t * 16 + ml) * LDK + kk + 2 * kh]);
            acc[t] = __builtin_amdgcn_wmma_f32_16x16x4_f32(
                false, a, false, b, (short)0, acc[t], false, false);
        }
    }

    // C/D layout: lanes 0-15 -> N=lane, rows M=r; lanes 16-31 -> N=lane-16, M=8+r
    const int rb = kh * 8;
#pragma unroll
    for (int t = 0; t < NT; ++t) {
#pragma unroll
        for (int r = 0; r < 8; ++r) {
            const int m = m0 + rb + r;
            if (m < n) H[(size_t)m * NOUT + t * 16 + ml] = acc[t][r];
        }
    }
}

// ---------------------------------------------------------------------------
// per-node attention dot products: as[i] = h_i . a_src ; ad[i] = h_i . a_dst
// ---------------------------------------------------------------------------
template <int D>
__global__ void node_dots(const float* __restrict__ H,
                          const float* __restrict__ a_s,
                          const float* __restrict__ a_d,
                          float* __restrict__ os, float* __restrict__ od, int n) {
    const int i = blockIdx.x * blockDim.x + threadIdx.x;
    if (i >= n) return;
    const float* __restrict__ h = H + (size_t)i * D;
    float ss = 0.0f, sd = 0.0f;
#pragma unroll
    for (int f = 0; f < D; ++f) {
        const float v = h[f];
        ss += v * a_s[f];
        sd += v * a_d[f];
    }
    os[i] = ss;
    od[i] = sd;
}

__global__ void init_neg_inf(float* __restrict__ m, int n) {
    const int i = blockIdx.x * blockDim.x + threadIdx.x;
    if (i < n) m[i] = -__builtin_inff();
}

// ---------------------------------------------------------------------------
// edge phase. edges [0,E) from edge_index; [E,ET) are self loops (s=d=e-E)
// ---------------------------------------------------------------------------
__device__ __forceinline__ void edge_sd(const int* __restrict__ ei, int E,
                                        unsigned e, int& s, int& d) {
    if (e < (unsigned)E) {
        s = ei[e];
        d = ei[(size_t)E + e];
    } else {
        s = d = (int)(e - (unsigned)E);
    }
}

__global__ void edge_logit_max(const int* __restrict__ ei, int E, int ET,
                               const float* __restrict__ as,
                               const float* __restrict__ ad,
                               float* __restrict__ mx) {
    const unsigned t = blockIdx.x * blockDim.x + threadIdx.x;
    if (t >= (unsigned)ET) return;
    int s, d;
    edge_sd(ei, E, t, s, d);
    atomicMaxF(&mx[d], lrelu(as[s] + ad[d], 0.2f));
}

__global__ void edge_exp_sum(const int* __restrict__ ei, int E, int ET,
                             const float* __restrict__ as,
                             const float* __restrict__ ad,
                             const float* __restrict__ mx,
                             float* __restrict__ ex, float* __restrict__ den) {
    const unsigned t = blockIdx.x * blockDim.x + threadIdx.x;
    if (t >= (unsigned)ET) return;
    int s, d;
    edge_sd(ei, E, t, s, d);
    const float v = __expf(lrelu(as[s] + ad[d], 0.2f) - mx[d]);
    ex[t] = v;
    atomicAdd(&den[d], v);
}

// one thread per (edge, feature): acc[d][f] += ex[e] * H[s][f]
template <int LOGD>
__global__ void edge_scatter(const int* __restrict__ ei, int E, int ET,
                             const float* __restrict__ H,
                             const float* __restrict__ ex,
                             float* __restrict__ acc) {
    constexpr unsigned D = 1u << LOGD;
    const unsigned t = blockIdx.x * blockDim.x + threadIdx.x;
    const unsigned e = t >> LOGD;
    if (e >= (unsigned)ET) return;
    const unsigned f = t & (D - 1u);
    int s, d;
    edge_sd(ei, E, e, s, d);
    atomicAdd(&acc[(size_t)d * D + f], ex[e] * H[(size_t)s * D + f]);
}

// out[i][f] = leaky_relu( acc[i][f] / (den[i] + 1e-16) + b[f], 0.01 )
template <int LOGD>
__global__ void node_finalize(const float* __restrict__ acc,
                              const float* __restrict__ den,
                              const float* __restrict__ b,
                              float* __restrict__ out, int n) {
    constexpr unsigned D = 1u << LOGD;
    const unsigned t = blockIdx.x * blockDim.x + threadIdx.x;
    if (t >= (unsigned)n * D) return;
    const unsigned i = t >> LOGD;
    const unsigned f = t & (D - 1u);
    const float v = acc[t] / (den[i] + 1e-16f) + b[f];
    out[t] = lrelu(v, 0.01f);
}

// ---------------------------------------------------------------------------
// launch
// ---------------------------------------------------------------------------
extern "C" void kernel_launch(void* const* d_in, const int* in_sizes, int n_in,
                              void* d_out, int out_size, void* d_ws, size_t ws_size,
                              hipStream_t stream) {
    const float* x   = (const float*)d_in[0];   // [n,64]
    const float* W1  = (const float*)d_in[1];   // [64,32]
    const float* aS1 = (const float*)d_in[2];
    const float* aD1 = (const float*)d_in[3];
    const float* b1  = (const float*)d_in[4];
    const float* W2  = (const float*)d_in[5];   // [32,64]
    const float* aS2 = (const float*)d_in[6];
    const float* aD2 = (const float*)d_in[7];
    const float* b2  = (const float*)d_in[8];
    const int*   ei  = (const int*)d_in[9];     // [2,E] (JAX x64-off -> int32)

    const int n  = in_sizes[0] / 64;
    const int E  = in_sizes[9] / 2;
    const int ET = E + n;                       // + self loops

    // workspace layout (floats); buffers reused across the two layers
    float* w    = (float*)d_ws;
    float* h    = w;                       // n*64 (layer1 uses first n*32)
    float* acc  = h + (size_t)n * 64;      // n*64 (layer1 uses first n*32)
    float* asN  = acc + (size_t)n * 64;    // n
    float* adN  = asN + (size_t)n;         // n
    float* mx   = adN + (size_t)n;         // n
    float* den  = mx + (size_t)n;          // n
    float* ex   = den + (size_t)n;         // ET

    float* midh    = (float*)d_out;                   // [n,32]
    float* decodex = (float*)d_out + (size_t)n * 32;  // [n,64]

    const int TPB = 256;
    const int tiles = (n + 15) / 16;
    const dim3 gemm_grid((tiles + 7) / 8);
    const dim3 node_grid((n + TPB - 1) / TPB);
    const dim3 edge_grid((ET + TPB - 1) / TPB);

    // ----------------------- layer 1 (64 -> 32) -----------------------
    hipMemsetAsync(acc, 0, (size_t)n * 32 * sizeof(float), stream);
    hipMemsetAsync(den, 0, (size_t)n * sizeof(float), stream);
    init_neg_inf<<<node_grid, TPB, 0, stream>>>(mx, n);

    gat_gemm_wmma<64, 32><<<gemm_grid, TPB, 0, stream>>>(x, W1, h, n);
    node_dots<32><<<node_grid, TPB, 0, stream>>>(h, aS1, aD1, asN, adN, n);

    edge_logit_max<<<edge_grid, TPB, 0, stream>>>(ei, E, ET, asN, adN, mx);
    edge_exp_sum<<<edge_grid, TPB, 0, stream>>>(ei, E, ET, asN, adN, mx, ex, den);
    {
        const size_t work = (size_t)ET * 32;
        edge_scatter<5><<<dim3((work + TPB - 1) / TPB), TPB, 0, stream>>>(
            ei, E, ET, h, ex, acc);
    }
    {
        const size_t work = (size_t)n * 32;
        node_finalize<5><<<dim3((work + TPB - 1) / TPB), TPB, 0, stream>>>(
            acc, den, b1, midh, n);
    }

    // ----------------------- layer 2 (32 -> 64) -----------------------
    hipMemsetAsync(acc, 0, (size_t)n * 64 * sizeof(float), stream);
    hipMemsetAsync(den, 0, (size_t)n * sizeof(float), stream);
    init_neg_inf<<<node_grid, TPB, 0, stream>>>(mx, n);

    gat_gemm_wmma<32, 64><<<gemm_grid, TPB, 0, stream>>>(midh, W2, h, n);
    node_dots<64><<<node_grid, TPB, 0, stream>>>(h, aS2, aD2, asN, adN, n);

    edge_logit_max<<<edge_grid, TPB, 0, stream>>>(ei, E, ET, asN, adN, mx);
    edge_exp_sum<<<edge_grid, TPB, 0, stream>>>(ei, E, ET, asN, adN, mx, ex, den);
    {
        const size_t work = (size_t)ET * 64;
        edge_scatter<6><<<dim3((work + TPB - 1) / TPB), TPB, 0, stream>>>(
            ei, E, ET, h, ex, acc);
    }
    {
        const size_t work = (size_t)n * 64;
        node_finalize<6><<<dim3((work + TPB - 1) / TPB), TPB, 0, stream>>>(
            acc, den, b2, decodex, n);
    }
}